// AttnHGCN_16140487098985
// MI455X (gfx1250) — compile-verified
//
#include <hip/hip_runtime.h>
#include <math.h>

#define CDIM 128

typedef __attribute__((ext_vector_type(2))) float v2f;
typedef __attribute__((ext_vector_type(8))) float v8f;

// Monotonic float<->uint mapping so atomicMax on unsigned == float max.
__device__ __forceinline__ unsigned orderKey(float f) {
  unsigned u = __float_as_uint(f);
  return (u & 0x80000000u) ? ~u : (u | 0x80000000u);
}
__device__ __forceinline__ float unorderKey(unsigned k) {
  unsigned u = (k & 0x80000000u) ? (k ^ 0x80000000u) : ~k;
  return __uint_as_float(u);
}

// ---------------------------------------------------------------------------
// P[M,128] = E[M,128] @ W[128,128] using V_WMMA_F32_16X16X4_F32.
// One 16x16 output tile per wave; 8 waves/block cover all 8 N-tiles of a row
// block. W_Q staged in LDS (64KB of the 320KB WGP pool).
// Fragment layouts per CDNA5 ISA 7.12.2:
//   A (16x4):  lanes 0-15 hold {K=0,K=1}, lanes 16-31 hold {K=2,K=3} for row M=lane&15
//   B (4x16):  lanes 0-15 hold rows {K=0,K=1}, lanes 16-31 rows {K=2,K=3}, col N=lane&15
//   C/D:       VGPR v -> row v + 8*(lane>=16), col lane&15
// ---------------------------------------------------------------------------
__global__ void __launch_bounds__(256) proj_wmma_kernel(
    const float* __restrict__ E, const float* __restrict__ W,
    float* __restrict__ P) {
  __shared__ float sW[CDIM * CDIM];
  for (int i = threadIdx.x; i < CDIM * CDIM / 4; i += 256)
    ((float4*)sW)[i] = ((const float4*)W)[i];
  __syncthreads();

  const int lane = threadIdx.x & 31;
  const int wave = threadIdx.x >> 5;
  const int half = lane >> 4;     // 0: K pair {0,1}, 1: K pair {2,3}
  const int lm   = lane & 15;
  const int m0   = blockIdx.x * 16;
  const int n0   = wave * 16;

  const float* arow = E + (size_t)(m0 + lm) * CDIM + 2 * half;
  const float* bcol = sW + (size_t)(2 * half) * CDIM + n0 + lm;

  v8f acc = {};
#pragma unroll
  for (int k0 = 0; k0 < CDIM; k0 += 4) {
    v2f a, b;
    a.x = arow[k0];
    a.y = arow[k0 + 1];
    b.x = bcol[k0 * CDIM];
    b.y = bcol[k0 * CDIM + CDIM];
    acc = __builtin_amdgcn_wmma_f32_16x16x4_f32(
        /*neg_a=*/false, a, /*neg_b=*/false, b,
        /*c_mod=*/(short)0, acc, /*reuse_a=*/false, /*reuse_b=*/false);
  }

#pragma unroll
  for (int v = 0; v < 8; ++v)
    P[(size_t)(m0 + v + 8 * half) * CDIM + n0 + lm] = acc[v];
}

// ---------------------------------------------------------------------------
// Pass 1: per-edge attention logits + segment max (wave per edge).
// score[e][h] = sum_d q[h,d] * k[h,d] * rel[h,d] / sqrt(64)
// lanes 0-15 own head 0 (channels 0-63), lanes 16-31 own head 1.
// ---------------------------------------------------------------------------
__global__ void __launch_bounds__(256) attn_score_kernel(
    const float* __restrict__ P, const float* __restrict__ relE,
    const int* __restrict__ head, const int* __restrict__ tail,
    const int* __restrict__ etype, float* __restrict__ escore,
    unsigned* __restrict__ smax, int nE) {
  const int e = blockIdx.x * 8 + (threadIdx.x >> 5);
  if (e >= nE) return;
  const int lane = threadIdx.x & 31;
  const int h = head[e], t = tail[e], r = etype[e] - 1;

  float4 q  = ((const float4*)(P + (size_t)h * CDIM))[lane];
  float4 k  = ((const float4*)(P + (size_t)t * CDIM))[lane];
  float4 rl = ((const float4*)(relE + (size_t)r * CDIM))[lane];
  float s = q.x * k.x * rl.x + q.y * k.y * rl.y +
            q.z * k.z * rl.z + q.w * k.w * rl.w;
  // reduce within each 16-lane half (one head per half)
  s += __shfl_xor(s, 1, 32);
  s += __shfl_xor(s, 2, 32);
  s += __shfl_xor(s, 4, 32);
  s += __shfl_xor(s, 8, 32);
  if ((lane & 15) == 0) {
    const int hd = lane >> 4;
    const float sc = s * 0.125f;  // 1/sqrt(D_K)=1/8
    escore[(size_t)e * 2 + hd] = sc;
    atomicMax(&smax[(size_t)h * 2 + hd], orderKey(sc));
  }
}

// ---------------------------------------------------------------------------
// Pass 2: w = exp(score - segmax); escore <- w; segment-sum via atomicAdd.
// ---------------------------------------------------------------------------
__global__ void __launch_bounds__(256) attn_norm_kernel(
    const int* __restrict__ head, float* __restrict__ escore,
    const unsigned* __restrict__ smax, float* __restrict__ ssum, int nE) {
  const int i = blockIdx.x * 256 + threadIdx.x;
  if (i >= 2 * nE) return;
  const int e = i >> 1, hd = i & 1;
  const int h = head[e];
  const float m = unorderKey(smax[(size_t)h * 2 + hd]);
  const float wv = expf(escore[i] - m);
  escore[i] = wv;
  atomicAdd(&ssum[(size_t)h * 2 + hd], wv);
}

// ---------------------------------------------------------------------------
// Pass 3: entity_agg[head] += (entity_emb[tail] * rel) * attn  (wave per edge)
// ---------------------------------------------------------------------------
__global__ void __launch_bounds__(256) attn_scatter_kernel(
    const float* __restrict__ curE, const float* __restrict__ relE,
    const int* __restrict__ head, const int* __restrict__ tail,
    const int* __restrict__ etype, const float* __restrict__ escore,
    const float* __restrict__ ssum, float* __restrict__ entAgg, int nE) {
  const int e = blockIdx.x * 8 + (threadIdx.x >> 5);
  if (e >= nE) return;
  const int lane = threadIdx.x & 31;
  const int h = head[e], t = tail[e], r = etype[e] - 1;
  const int hd = lane >> 4;  // head id for this lane's channels
  const float a = escore[(size_t)e * 2 + hd] /
                  fmaxf(ssum[(size_t)h * 2 + hd], 1e-16f);
  float4 v  = ((const float4*)(curE + (size_t)t * CDIM))[lane];
  float4 rl = ((const float4*)(relE + (size_t)r * CDIM))[lane];
  float* dst = entAgg + (size_t)h * CDIM + lane * 4;
  atomicAdd(dst + 0, v.x * rl.x * a);
  atomicAdd(dst + 1, v.y * rl.y * a);
  atomicAdd(dst + 2, v.z * rl.z * a);
  atomicAdd(dst + 3, v.w * rl.w * a);
}

// ---------------------------------------------------------------------------
// user_agg[u] += w * entity_emb[item]  (wave per interaction edge)
// ---------------------------------------------------------------------------
__global__ void __launch_bounds__(256) user_scatter_kernel(
    const float* __restrict__ curE, const int* __restrict__ iu,
    const int* __restrict__ ii, const float* __restrict__ iw,
    float* __restrict__ userAgg, int nE) {
  const int e = blockIdx.x * 8 + (threadIdx.x >> 5);
  if (e >= nE) return;
  const int lane = threadIdx.x & 31;
  const int u = iu[e], it = ii[e];
  const float w = iw[e];
  float4 v = ((const float4*)(curE + (size_t)it * CDIM))[lane];
  float* dst = userAgg + (size_t)u * CDIM + lane * 4;
  atomicAdd(dst + 0, v.x * w);
  atomicAdd(dst + 1, v.y * w);
  atomicAdd(dst + 2, v.z * w);
  atomicAdd(dst + 3, v.w * w);
}

// ---------------------------------------------------------------------------
// Row-wise L2 normalize; optionally emit normalized row (next hop's input)
// and accumulate into residual. Wave per row.
// ---------------------------------------------------------------------------
__global__ void __launch_bounds__(256) norm_res_kernel(
    const float* __restrict__ agg, float* __restrict__ normed,
    float* __restrict__ res, int nrows) {
  const int row = blockIdx.x * 8 + (threadIdx.x >> 5);
  if (row >= nrows) return;
  const int lane = threadIdx.x & 31;
  float4 v = ((const float4*)(agg + (size_t)row * CDIM))[lane];
  float ss = v.x * v.x + v.y * v.y + v.z * v.z + v.w * v.w;
#pragma unroll
  for (int o = 16; o > 0; o >>= 1) ss += __shfl_xor(ss, o, 32);
  const float inv = 1.0f / fmaxf(sqrtf(ss), 1e-12f);
  float4 n = make_float4(v.x * inv, v.y * inv, v.z * inv, v.w * inv);
  if (normed) ((float4*)(normed + (size_t)row * CDIM))[lane] = n;
  float4* rp = (float4*)(res + (size_t)row * CDIM) + lane;
  float4 rc = *rp;
  rc.x += n.x; rc.y += n.y; rc.z += n.z; rc.w += n.w;
  *rp = rc;
}

// ---------------------------------------------------------------------------
extern "C" void kernel_launch(void* const* d_in, const int* in_sizes, int n_in,
                              void* d_out, int out_size, void* d_ws,
                              size_t ws_size, hipStream_t stream) {
  (void)n_in; (void)out_size; (void)ws_size;
  const float* user0 = (const float*)d_in[0];
  const float* ent0  = (const float*)d_in[1];
  const int*   eidx  = (const int*)d_in[2];
  const int*   etype = (const int*)d_in[3];
  const int*   inter = (const int*)d_in[4];
  const float* iw    = (const float*)d_in[5];
  const float* relE  = (const float*)d_in[6];
  const float* WQ    = (const float*)d_in[7];

  const int nU   = in_sizes[0] / CDIM;   // 100000
  const int nEnt = in_sizes[1] / CDIM;   // 200000
  const int eKG  = in_sizes[3];          // 1,000,000
  const int eUI  = in_sizes[5];          // 1,000,000
  const int* head = eidx;
  const int* tail = eidx + eKG;
  const int* iu   = inter;
  const int* ii   = inter + eUI;

  float* entRes  = (float*)d_out;
  float* userRes = entRes + (size_t)nEnt * CDIM;

  // Workspace layout (P is reused as the entity-agg buffer after the score
  // pass consumes it): total ~267 MB.
  char* w = (char*)d_ws;
  float*    P      = (float*)w;    w += (size_t)nEnt * CDIM * sizeof(float);
  float*    entA   = (float*)w;    w += (size_t)nEnt * CDIM * sizeof(float);
  float*    userA  = (float*)w;    w += (size_t)nU * CDIM * sizeof(float);
  float*    escore = (float*)w;    w += (size_t)eKG * 2 * sizeof(float);
  float*    ssum   = (float*)w;    w += (size_t)nEnt * 2 * sizeof(float);
  unsigned* smax   = (unsigned*)w;

  // residual init: res = input embeddings
  hipMemcpyAsync(entRes, ent0, (size_t)nEnt * CDIM * sizeof(float),
                 hipMemcpyDeviceToDevice, stream);
  hipMemcpyAsync(userRes, user0, (size_t)nU * CDIM * sizeof(float),
                 hipMemcpyDeviceToDevice, stream);

  const float* cur = ent0;
  for (int hop = 0; hop < 2; ++hop) {
    hipMemsetAsync(ssum, 0, (size_t)nEnt * 2 * sizeof(float), stream);
    hipMemsetAsync(smax, 0, (size_t)nEnt * 2 * sizeof(unsigned), stream);
    hipMemsetAsync(userA, 0, (size_t)nU * CDIM * sizeof(float), stream);

    // P = cur @ W_Q  (WMMA)
    proj_wmma_kernel<<<nEnt / 16, 256, 0, stream>>>(cur, WQ, P);
    // attention logits + segment max
    attn_score_kernel<<<(eKG + 7) / 8, 256, 0, stream>>>(
        P, relE, head, tail, etype, escore, smax, eKG);
    // exp + segment sum
    attn_norm_kernel<<<(2 * eKG + 255) / 256, 256, 0, stream>>>(
        head, escore, smax, ssum, eKG);
    // P no longer needed -> reuse as entity aggregation buffer
    hipMemsetAsync(P, 0, (size_t)nEnt * CDIM * sizeof(float), stream);
    attn_scatter_kernel<<<(eKG + 7) / 8, 256, 0, stream>>>(
        cur, relE, head, tail, etype, escore, ssum, P, eKG);
    // user aggregation
    user_scatter_kernel<<<(eUI + 7) / 8, 256, 0, stream>>>(
        cur, iu, ii, iw, userA, eUI);
    // normalize + residual; entA becomes next hop's entity input
    norm_res_kernel<<<(nEnt + 7) / 8, 256, 0, stream>>>(
        P, hop == 0 ? entA : (float*)nullptr, entRes, nEnt);
    norm_res_kernel<<<(nU + 7) / 8, 256, 0, stream>>>(
        userA, (float*)nullptr, userRes, nU);
    cur = entA;
  }
}